// SpatialEncoder_5935644803789
// MI455X (gfx1250) — compile-verified
//
#include <hip/hip_runtime.h>
#include <stdint.h>

// GraphSAGE 2-layer inference for MI455X (gfx1250, wave32).
// Edge phase is L2-resident scatter/gather (working set ~60MB << 192MB L2);
// dense phase uses V_WMMA_F32_16X16X4_F32 (exact fp32, GEMM is tiny vs edges).

typedef float v2f __attribute__((ext_vector_type(2)));
typedef float v8f __attribute__((ext_vector_type(8)));

#define D 128   // feature dim (all layers)
#define KC 32   // K-chunk staged in LDS per block iteration

__global__ void zero_f32(float* __restrict__ p, long n) {
  long i = (long)blockIdx.x * blockDim.x + threadIdx.x;
  long stride = (long)gridDim.x * blockDim.x;
  for (; i < n; i += stride) p[i] = 0.0f;
}

__global__ void degree_kernel(const long long* __restrict__ dst,
                              float* __restrict__ deg, int E) {
  int e = blockIdx.x * blockDim.x + threadIdx.x;
  if (e < E) atomicAdd(&deg[(int)dst[e]], 1.0f);
}

__global__ void invdeg_kernel(float* __restrict__ deg, int N) {
  int i = blockIdx.x * blockDim.x + threadIdx.x;
  if (i < N) deg[i] = 1.0f / fmaxf(deg[i], 1.0f);
}

// One wave32 per edge: lane l moves features [4l..4l+3] of x[src] into
// agg[dst] via 4 global f32 atomics (no-return -> STOREcnt path).
__global__ void scatter_kernel(const float* __restrict__ x,
                               const long long* __restrict__ src,
                               const long long* __restrict__ dst,
                               float* __restrict__ agg, int E) {
  int wave = (blockIdx.x * blockDim.x + threadIdx.x) >> 5;
  int lane = threadIdx.x & 31;
  if (wave >= E) return;
  int s = (int)src[wave];
  int d = (int)dst[wave];
  const float4 v = *(const float4*)(x + (long)s * D + lane * 4);
  float* a = agg + (long)d * D + lane * 4;
  atomicAdd(a + 0, v.x);
  atomicAdd(a + 1, v.y);
  atomicAdd(a + 2, v.z);
  atomicAdd(a + 3, v.w);
}

// out[r,:] = act( (A[r,:]*invdeg[r]) @ Wl + bias + X[r,:] @ Wr )
// Block = 256 threads = 8 waves; each wave owns a 16-row x 128-col strip.
// Weight K-panels (KC x 128, both matrices) staged in LDS transposed with
// stride KC+1 to avoid bank conflicts. WMMA: f32 16x16x4.
__global__ __launch_bounds__(256)
void gemm_kernel(const float* __restrict__ A,      // aggregated [N,D]
                 const float* __restrict__ X,      // skip input [N,D]
                 const float* __restrict__ Wl,     // [D,D]
                 const float* __restrict__ Wr,     // [D,D]
                 const float* __restrict__ bias,   // [D]
                 const float* __restrict__ invdeg, // [N]
                 float* __restrict__ out,          // [N,D]
                 int N, int relu) {
  __shared__ float lWl[D * (KC + 1)];
  __shared__ float lWr[D * (KC + 1)];

  const int tid  = threadIdx.x;
  const int wave = tid >> 5;
  const int lane = tid & 31;
  const int half = lane >> 4;   // 0: lanes 0-15, 1: lanes 16-31
  const int lrow = lane & 15;

  const int r0 = blockIdx.x * 128 + wave * 16;

  // A-fragment row for this lane; clamp so EXEC stays all-1s for WMMA
  int arow = r0 + lrow;
  if (arow >= N) arow = N - 1;
  const float scale = invdeg[arow];

  v8f acc[8];
  const v8f vzero = {0.f, 0.f, 0.f, 0.f, 0.f, 0.f, 0.f, 0.f};
  for (int t = 0; t < 8; ++t) acc[t] = vzero;

  for (int kc = 0; kc < D; kc += KC) {
    __syncthreads();
    // Stage Wl/Wr[kc:kc+KC, :] transposed: lW[col*(KC+1) + r].
    // Global reads coalesced (consecutive tid -> consecutive col).
    for (int idx = tid; idx < KC * D; idx += 256) {
      int r = idx >> 7;        // 0..KC-1
      int c = idx & (D - 1);   // 0..127
      lWl[c * (KC + 1) + r] = Wl[(kc + r) * D + c];
      lWr[c * (KC + 1) + r] = Wr[(kc + r) * D + c];
    }
    __syncthreads();

    for (int k0 = 0; k0 < KC; k0 += 4) {
      // A fragment (16x4): lanes 0-15 hold K=k,k+1; lanes 16-31 hold K=k+2,k+3
      const int kk = kc + k0 + 2 * half;
      v2f aL, aX;
      aL.x = A[(long)arow * D + kk] * scale;
      aL.y = A[(long)arow * D + kk + 1] * scale;
      aX.x = X[(long)arow * D + kk];
      aX.y = X[(long)arow * D + kk + 1];

      for (int nt = 0; nt < 8; ++nt) {
        const int col = nt * 16 + lrow;
        // B fragment (4x16): lane holds column `col`, same K split as A
        v2f bL, bR;
        const int kb = col * (KC + 1) + k0 + 2 * half;
        bL.x = lWl[kb];
        bL.y = lWl[kb + 1];
        bR.x = lWr[kb];
        bR.y = lWr[kb + 1];
        // D = A*B + C  (8 args: neg_a, A, neg_b, B, c_mod, C, reuse_a, reuse_b)
        acc[nt] = __builtin_amdgcn_wmma_f32_16x16x4_f32(
            false, aL, false, bL, (short)0, acc[nt], false, false);
        acc[nt] = __builtin_amdgcn_wmma_f32_16x16x4_f32(
            false, aX, false, bR, (short)0, acc[nt], false, false);
      }
    }
  }

  // Epilogue: C/D layout -> VGPR j holds row j (lanes 0-15) / row 8+j (16-31)
  for (int nt = 0; nt < 8; ++nt) {
    const int col = nt * 16 + lrow;
    const float b = bias[col];
    for (int j = 0; j < 8; ++j) {
      int row = r0 + j + 8 * half;
      if (row < N) {
        float v = acc[nt][j] + b;
        if (relu) v = fmaxf(v, 0.0f);
        out[(long)row * D + col] = v;
      }
    }
  }
}

extern "C" void kernel_launch(void* const* d_in, const int* in_sizes, int n_in,
                              void* d_out, int out_size, void* d_ws, size_t ws_size,
                              hipStream_t stream) {
  const float*     x   = (const float*)d_in[0];
  const long long* ei  = (const long long*)d_in[1];   // int64 [2,E]
  const float*     wl1 = (const float*)d_in[2];
  const float*     bl1 = (const float*)d_in[3];
  const float*     wr1 = (const float*)d_in[4];
  const float*     wl2 = (const float*)d_in[5];
  const float*     bl2 = (const float*)d_in[6];
  const float*     wr2 = (const float*)d_in[7];

  const int N = in_sizes[0] / D;
  const int E = in_sizes[1] / 2;
  const long long* src = ei;       // row 0
  const long long* dst = ei + E;   // row 1

  // workspace: deg[N] | agg[N*D] | h[N*D]   (agg reused for layer 2)
  float* deg = (float*)d_ws;
  float* agg = deg + N;
  float* h   = agg + (long)N * D;
  float* out = (float*)d_out;

  const int edgeBlocks  = (E + 255) / 256;
  const int nodeBlocks  = (N + 255) / 256;
  const int gemmBlocks  = (N + 127) / 128;
  const int waveBlocks  = (E * 32 + 255) / 256;   // one wave per edge

  // ---- shared preprocessing ----
  zero_f32<<<2048, 256, 0, stream>>>(deg, (long)N * (D + 1));        // deg+agg
  degree_kernel<<<edgeBlocks, 256, 0, stream>>>(dst, deg, E);
  invdeg_kernel<<<nodeBlocks, 256, 0, stream>>>(deg, N);             // deg := 1/max(deg,1)

  // ---- layer 1 ----
  scatter_kernel<<<waveBlocks, 256, 0, stream>>>(x, src, dst, agg, E);
  gemm_kernel<<<gemmBlocks, 256, 0, stream>>>(agg, x, wl1, bl1, wr1, deg, h, N, 1);

  // ---- layer 2 (reuse agg) ----
  zero_f32<<<2048, 256, 0, stream>>>(agg, (long)N * D);
  scatter_kernel<<<waveBlocks, 256, 0, stream>>>(h, src, dst, agg, E);
  gemm_kernel<<<gemmBlocks, 256, 0, stream>>>(agg, h, wl2, bl2, wr2, deg, out, N, 0);
}